// FirstLayer_21973052686650
// MI455X (gfx1250) — compile-verified
//
#include <hip/hip_runtime.h>
#include <cstdint>

// Problem constants (match reference)
constexpr int   N    = 50000;
constexpr long  E    = 1600000;
constexpr int   F_IN = 128;
constexpr int   F_E  = 32;
constexpr int   H    = 4;
constexpr int   C    = 32;
constexpr int   HC   = H * C;          // 128
constexpr long  EN   = E + (long)N;    // edges incl. self loops

typedef float v2f __attribute__((ext_vector_type(2)));
typedef float v8f __attribute__((ext_vector_type(8)));

// ---------- ordered-float <-> uint monotone encoding for atomicMax ----------
__device__ __forceinline__ unsigned fenc(float f) {
    unsigned u = __float_as_uint(f);
    return (u & 0x80000000u) ? ~u : (u | 0x80000000u);
}
__device__ __forceinline__ float fdec(unsigned k) {
    return __uint_as_float((k & 0x80000000u) ? (k ^ 0x80000000u) : ~k);
}

// ---------------- kernel 0: init out=bias, denom=0, amax=key0, ea_sum=0 -----
__global__ void k_init(const float* __restrict__ bias, float* __restrict__ out,
                       float* __restrict__ denom, unsigned* __restrict__ amax,
                       float* __restrict__ ea_sum) {
    long i = (long)blockIdx.x * blockDim.x + threadIdx.x;
    if (i < (long)N * HC) out[i] = bias[i & (HC - 1)];
    if (i < (long)N * H) { denom[i] = 0.0f; amax[i] = 0u; }
    if (i < F_E) ea_sum[i] = 0.0f;
}

// ---------------- kernel 1: column sums of edge_attr ------------------------
__global__ void k_ea_colsum(const float* __restrict__ edge_attr, float* __restrict__ ea_sum) {
    int g = blockIdx.x * blockDim.x + threadIdx.x;
    int col = g & (F_E - 1);
    long row0 = g >> 5;
    long stride = ((long)gridDim.x * blockDim.x) >> 5;
    float s = 0.0f;
    for (long r = row0; r < E; r += stride) s += edge_attr[r * F_E + col];
    atomicAdd(&ea_sum[col], s);
}

// ------- kernel 2: V[k][h] = sum_c W_e[k, h*C+c]*att_edge[h,c]; ea_mean -----
__global__ void k_prep(const float* __restrict__ W_e, const float* __restrict__ att_edge,
                       const float* __restrict__ ea_sum,
                       float* __restrict__ Vbuf, float* __restrict__ ea_mean) {
    int t = threadIdx.x;           // 0..127 ; t = k*H + h
    int k = t >> 2, h = t & 3;
    float s = 0.0f;
    for (int c = 0; c < C; ++c) s += W_e[k * HC + h * C + c] * att_edge[h * C + c];
    Vbuf[t] = s;
    if (t < F_E) ea_mean[t] = ea_sum[t] * (1.0f / (float)E);
}

// ---------------- kernel 3: xl = x @ W  via f32 WMMA ------------------------
// block = 256 threads (8 waves); wave w computes rows [blk*128 + w*16, +16) x all 128 cols.
// W is staged TRANSPOSED in LDS (pitch 130) so each lane's B fragment
// {W[ka][col], W[ka+1][col]} is one aligned 8-byte ds_load_b64, bank-conflict-free.
constexpr int LP = F_IN + 2;   // padded pitch in floats
__global__ void k_gemm_wmma(const float* __restrict__ x, const float* __restrict__ W,
                            float* __restrict__ xl) {
    __shared__ float Wt[HC * LP];          // 66.5 KB of the 320 KB WGP LDS
    for (int idx = threadIdx.x; idx < F_IN * HC; idx += 256) {
        int k = idx >> 7;          // row of W (K index)
        int c = idx & 127;         // col of W (N index)
        Wt[c * LP + k] = W[idx];   // contiguous global reads, transposed LDS write
    }
    __syncthreads();

    const int wave   = threadIdx.x >> 5;
    const int lane   = threadIdx.x & 31;
    const int lane16 = lane & 15;
    const int half   = lane >> 4;          // 0: K pair {0,1}, 1: K pair {2,3}
    const int m0     = blockIdx.x * 128 + wave * 16;

    int rowA = m0 + lane16;
    if (rowA > N - 1) rowA = N - 1;        // clamp loads; stores predicated later
    const float* xrow = x + (long)rowA * F_IN;

    v8f acc[8];
#pragma unroll
    for (int t = 0; t < 8; ++t) acc[t] = (v8f){0,0,0,0,0,0,0,0};

    for (int kk = 0; kk < F_IN; kk += 4) {
        const int ka = kk + half * 2;
        float2 av = *(const float2*)(xrow + ka);      // A frag: row m, K=ka..ka+1
        v2f a; a.x = av.x; a.y = av.y;
#pragma unroll
        for (int t = 0; t < 8; ++t) {
            const int col = t * 16 + lane16;
            float2 bv = *(const float2*)(Wt + col * LP + ka);  // single ds_load_b64
            v2f b; b.x = bv.x; b.y = bv.y;
            acc[t] = __builtin_amdgcn_wmma_f32_16x16x4_f32(
                false, a, false, b, (short)0, acc[t], false, false);
        }
    }
    // D layout: VGPR r -> row m0 + r + 8*half, col = t*16 + lane16
#pragma unroll
    for (int r = 0; r < 8; ++r) {
        const int row = m0 + r + half * 8;
        if (row < N) {
#pragma unroll
            for (int t = 0; t < 8; ++t)
                xl[(long)row * HC + t * 16 + lane16] = acc[t][r];
        }
    }
}

// ---------------- kernel 4: a_src / a_dst per (node, head) ------------------
__global__ void k_node_att(const float* __restrict__ xl, const float* __restrict__ att_src,
                           const float* __restrict__ att_dst,
                           float* __restrict__ a_src, float* __restrict__ a_dst) {
    int t = blockIdx.x * blockDim.x + threadIdx.x;
    if (t >= N * H) return;
    int nd = t >> 2, h = t & 3;
    const float* xr = xl + (long)nd * HC + h * C;
    const float* as = att_src + h * C;
    const float* ad = att_dst + h * C;
    float s1 = 0.0f, s2 = 0.0f;
#pragma unroll 8
    for (int c = 0; c < C; ++c) { float v = xr[c]; s1 += v * as[c]; s2 += v * ad[c]; }
    a_src[t] = s1; a_dst[t] = s2;
}

// --------- kernel 5 (pass A): alpha_raw + segment max over dst --------------
__global__ void k_edge_alpha(const long long* __restrict__ ei, const float* __restrict__ edge_attr,
                             const float* __restrict__ a_src, const float* __restrict__ a_dst,
                             const float* __restrict__ Vbuf, const float* __restrict__ ea_mean,
                             float* __restrict__ alpha, unsigned* __restrict__ amax) {
    __shared__ float Vs[F_E * H];   // 128
    __shared__ float Ms[F_E];       // 32
    if (threadIdx.x < F_E * H) Vs[threadIdx.x] = Vbuf[threadIdx.x];
    if (threadIdx.x < F_E)     Ms[threadIdx.x] = ea_mean[threadIdx.x];
    __syncthreads();

    long e = (long)blockIdx.x * blockDim.x + threadIdx.x;
    if (e >= EN) return;
    int s, d;
    const float* ea;
    if (e < E) { s = (int)ei[e]; d = (int)ei[E + e]; ea = edge_attr + e * F_E; }
    else       { s = d = (int)(e - E); ea = Ms; }

    float ae0 = 0, ae1 = 0, ae2 = 0, ae3 = 0;
#pragma unroll 8
    for (int k = 0; k < F_E; ++k) {
        float v = ea[k];
        ae0 += v * Vs[k * 4 + 0]; ae1 += v * Vs[k * 4 + 1];
        ae2 += v * Vs[k * 4 + 2]; ae3 += v * Vs[k * 4 + 3];
    }
    float aeh[4] = {ae0, ae1, ae2, ae3};
#pragma unroll
    for (int h = 0; h < H; ++h) {
        float a = a_src[(long)s * H + h] + a_dst[(long)d * H + h] + aeh[h];
        a = (a > 0.0f) ? a : 0.2f * a;                 // leaky_relu(0.2)
        alpha[e * H + h] = a;
        atomicMax(&amax[(long)d * H + h], fenc(a));
    }
}

// --------- kernel 6 (pass B): exp(alpha - max), segment sum -----------------
__global__ void k_edge_norm(const long long* __restrict__ ei, float* __restrict__ alpha,
                            const unsigned* __restrict__ amax, float* __restrict__ denom) {
    long e = (long)blockIdx.x * blockDim.x + threadIdx.x;
    if (e >= EN) return;
    int d = (e < E) ? (int)ei[E + e] : (int)(e - E);
    float4 a = *(const float4*)(alpha + e * H);
    float w0 = __expf(a.x - fdec(amax[(long)d * H + 0]));
    float w1 = __expf(a.y - fdec(amax[(long)d * H + 1]));
    float w2 = __expf(a.z - fdec(amax[(long)d * H + 2]));
    float w3 = __expf(a.w - fdec(amax[(long)d * H + 3]));
    *(float4*)(alpha + e * H) = make_float4(w0, w1, w2, w3);
    atomicAdd(&denom[(long)d * H + 0], w0);
    atomicAdd(&denom[(long)d * H + 1], w1);
    atomicAdd(&denom[(long)d * H + 2], w2);
    atomicAdd(&denom[(long)d * H + 3], w3);
}

// --------- kernel 7 (pass C): out[dst] += alpha_hat * xl[src] ---------------
// one wave (32 lanes) per edge; lane = channel c, loop over heads
__global__ void k_scatter(const long long* __restrict__ ei, const float* __restrict__ xl,
                          const float* __restrict__ alpha, const float* __restrict__ denom,
                          float* __restrict__ out) {
    const int lane = threadIdx.x & 31;
    long e = (long)blockIdx.x * (blockDim.x >> 5) + (threadIdx.x >> 5);
    if (e >= EN) return;
    int s, d;
    if (e < E) { s = (int)ei[e]; d = (int)ei[E + e]; }
    else       { s = d = (int)(e - E); }
    const float* xr = xl + (long)s * HC;
    float* orow = out + (long)d * HC;
#pragma unroll
    for (int h = 0; h < H; ++h) {
        float w = alpha[e * H + h] / (denom[(long)d * H + h] + 1e-16f);
        atomicAdd(&orow[h * C + lane], w * xr[h * C + lane]);
    }
}

// ---------------------------------------------------------------------------
extern "C" void kernel_launch(void* const* d_in, const int* in_sizes, int n_in,
                              void* d_out, int out_size, void* d_ws, size_t ws_size,
                              hipStream_t stream) {
    const float*     x        = (const float*)d_in[0];
    const long long* ei       = (const long long*)d_in[1];   // int64 [2, E]
    const float*     eattr    = (const float*)d_in[2];
    const float*     W        = (const float*)d_in[3];
    const float*     att_src  = (const float*)d_in[4];
    const float*     att_dst  = (const float*)d_in[5];
    const float*     W_e      = (const float*)d_in[6];
    const float*     att_edge = (const float*)d_in[7];
    const float*     bias     = (const float*)d_in[8];
    float*           out      = (float*)d_out;

    // workspace layout (floats)  ~55 MB total
    float*    ws      = (float*)d_ws;
    float*    xl      = ws;                              // N*128
    float*    a_src   = xl + (long)N * HC;               // N*H
    float*    a_dst   = a_src + (long)N * H;             // N*H
    unsigned* amax    = (unsigned*)(a_dst + (long)N * H);// N*H
    float*    denom   = (float*)(amax + (long)N * H);    // N*H
    float*    alpha   = denom + (long)N * H;             // (E+N)*H
    float*    Vbuf    = alpha + EN * H;                  // 128
    float*    ea_sum  = Vbuf + F_E * H;                  // 32
    float*    ea_mean = ea_sum + F_E;                    // 32

    const int TB = 256;

    // 0) init
    {
        long total = (long)N * HC;
        k_init<<<(int)((total + TB - 1) / TB), TB, 0, stream>>>(bias, out, denom, amax, ea_sum);
    }
    // 1) edge_attr column sums
    k_ea_colsum<<<128, TB, 0, stream>>>(eattr, ea_sum);
    // 2) collapsed edge projection V and mean fill
    k_prep<<<1, 128, 0, stream>>>(W_e, att_edge, ea_sum, Vbuf, ea_mean);
    // 3) xl = x @ W   (f32 WMMA)
    k_gemm_wmma<<<(N + 127) / 128, TB, 0, stream>>>(x, W, xl);
    // 4) node attention scalars
    k_node_att<<<(N * H + TB - 1) / TB, TB, 0, stream>>>(xl, att_src, att_dst, a_src, a_dst);
    // 5) pass A: alpha_raw + segment max
    k_edge_alpha<<<(int)((EN + TB - 1) / TB), TB, 0, stream>>>(ei, eattr, a_src, a_dst,
                                                              Vbuf, ea_mean, alpha, amax);
    // 6) pass B: exp + segment sum
    k_edge_norm<<<(int)((EN + TB - 1) / TB), TB, 0, stream>>>(ei, alpha, amax, denom);
    // 7) pass C: weighted scatter-add
    {
        const int edgesPerBlock = TB / 32;
        k_scatter<<<(int)((EN + edgesPerBlock - 1) / edgesPerBlock), TB, 0, stream>>>(
            ei, xl, alpha, denom, out);
    }
}